// GlobalAttentionPooling_16458314678922
// MI455X (gfx1250) — compile-verified
//
#include <hip/hip_runtime.h>
#include <hip/hip_bf16.h>
#include <math.h>

// Problem constants from the reference setup: D = H = 256 (hardcoded in tiling).
#define DFEAT 256

typedef __attribute__((ext_vector_type(2))) float v2f;
typedef __attribute__((ext_vector_type(8))) float v8f;

// ---------------------------------------------------------------------------
// Kernel 1: segment boundaries via lower_bound on the sorted segment_ids.
// start[b] = first row with id >= b ; start[B] = N.
// ---------------------------------------------------------------------------
__global__ void gap_bounds_kernel(const int* __restrict__ ids,
                                  int* __restrict__ start, int N, int B) {
    int b = blockIdx.x * blockDim.x + threadIdx.x;
    if (b > B) return;
    if (b == B) { start[B] = N; return; }
    int lo = 0, hi = N;
    while (lo < hi) {
        int mid = (lo + hi) >> 1;
        if (ids[mid] < b) lo = mid + 1; else hi = mid;
    }
    start[b] = lo;
}

// ---------------------------------------------------------------------------
// Kernel 2: gate GEMV  g[i] = feat[i,:] . Wg   (one wave32 per row, float4 IO)
// ---------------------------------------------------------------------------
__global__ void gap_gate_kernel(const float* __restrict__ feat,
                                const float* __restrict__ Wg,
                                float* __restrict__ g, int N) {
    int wave = (int)((blockIdx.x * (size_t)blockDim.x + threadIdx.x) >> 5);
    int lane = threadIdx.x & 31;
    if (wave >= N) return;
    const float4* fr = (const float4*)(feat + (size_t)wave * DFEAT);
    const float4* wg = (const float4*)Wg;
    float4 a0 = fr[lane * 2 + 0];
    float4 a1 = fr[lane * 2 + 1];
    float4 w0 = wg[lane * 2 + 0];
    float4 w1 = wg[lane * 2 + 1];
    float s = a0.x * w0.x + a0.y * w0.y + a0.z * w0.z + a0.w * w0.w
            + a1.x * w1.x + a1.y * w1.y + a1.z * w1.z + a1.w * w1.w;
    #pragma unroll
    for (int off = 16; off > 0; off >>= 1)
        s += __shfl_xor(s, off, 32);
    if (lane == 0) g[wave] = s;
}

// ---------------------------------------------------------------------------
// Kernel 3: per-segment softmax-weighted feature sum.
//   S[b,t] = (sum_r exp(g[r]-m) * feat[r,t]) / (sum_r exp(g[r]-m))
// One 256-thread block per segment; rows of the segment are contiguous.
// ---------------------------------------------------------------------------
__global__ void gap_segment_kernel(const float* __restrict__ feat,
                                   const float* __restrict__ g,
                                   const int* __restrict__ start,
                                   float* __restrict__ S) {
    const int b = blockIdx.x;
    const int t = threadIdx.x;                 // 0..255 (== feature column)
    const int s0 = start[b], s1 = start[b + 1];

    __shared__ float red[256];
    __shared__ float e_lds[256];

    if (s0 >= s1) {                            // empty segment -> zero row
        S[(size_t)b * DFEAT + t] = 0.0f;
        return;
    }

    // pass 1: segment max of g
    float m = -INFINITY;
    for (int r = s0 + t; r < s1; r += 256) m = fmaxf(m, g[r]);
    red[t] = m; __syncthreads();
    #pragma unroll
    for (int off = 128; off > 0; off >>= 1) {
        if (t < off) red[t] = fmaxf(red[t], red[t + off]);
        __syncthreads();
    }
    m = red[0];
    __syncthreads();

    // pass 2: chunked exp into LDS + coalesced weighted accumulation
    float acc = 0.0f, dpart = 0.0f;
    for (int c0 = s0; c0 < s1; c0 += 256) {
        int cn = s1 - c0; if (cn > 256) cn = 256;
        if (t < cn) {
            float e = __expf(g[c0 + t] - m);
            e_lds[t] = e;
            dpart += e;
        }
        __syncthreads();
        // prefetch next chunk's column while we chew on this one
        if (c0 + 256 < s1)
            __builtin_prefetch(&feat[(size_t)(c0 + 256) * DFEAT + t], 0, 1);
        for (int j = 0; j < cn; ++j)
            acc = fmaf(e_lds[j], feat[(size_t)(c0 + j) * DFEAT + t], acc);
        __syncthreads();
    }

    // reduce denom, normalize, write S row
    red[t] = dpart; __syncthreads();
    #pragma unroll
    for (int off = 128; off > 0; off >>= 1) {
        if (t < off) red[t] += red[t + off];
        __syncthreads();
    }
    S[(size_t)b * DFEAT + t] = acc / red[0];
}

// ---------------------------------------------------------------------------
// Kernel 4: out = S @ Wf + bf  via V_WMMA_F32_16X16X4_F32.
// M = B (4096), N = H (256), K = D (256). One wave per 16x16 output tile,
// 64 WMMA steps along K. A/B VGPR layout per CDNA5 ISA 16x4 f32 tables:
//   lanes 0-15: row/col = lane,    vgpr0=K0, vgpr1=K1
//   lanes16-31: row/col = lane-16, vgpr0=K2, vgpr1=K3
// Bias is added at store time only for non-empty segments (reference yields
// exact zeros for empty segments).
// ---------------------------------------------------------------------------
__global__ void gap_gemm_kernel(const float* __restrict__ S,
                                const float* __restrict__ Wf,
                                const float* __restrict__ bf,
                                const int* __restrict__ start,
                                float* __restrict__ out, int nTilesN) {
    const int lane = threadIdx.x & 31;
    const int waveInBlk = threadIdx.x >> 5;
    const int tile = blockIdx.x * (blockDim.x >> 5) + waveInBlk;
    const int mTile = tile / nTilesN;
    const int nTile = tile % nTilesN;
    const int m0 = mTile * 16, n0 = nTile * 16;
    const int half = lane >> 4;        // 0: K{0,1}   1: K{2,3}
    const int l = lane & 15;           // row (A) / col (B)
    const int kk = half * 2;

    v8f c = {};
    const float* Arow = S + (size_t)(m0 + l) * DFEAT;
    for (int k0 = 0; k0 < DFEAT; k0 += 4) {
        float2 apair = *(const float2*)(Arow + k0 + kk);
        v2f a; a.x = apair.x; a.y = apair.y;
        v2f bb;
        bb.x = Wf[(size_t)(k0 + kk + 0) * DFEAT + n0 + l];
        bb.y = Wf[(size_t)(k0 + kk + 1) * DFEAT + n0 + l];
        c = __builtin_amdgcn_wmma_f32_16x16x4_f32(
                /*neg_a=*/false, a, /*neg_b=*/false, bb,
                /*c_mod=*/(short)0, c, /*reuse_a=*/false, /*reuse_b=*/false);
    }

    const float bias = bf[n0 + l];
    #pragma unroll
    for (int r = 0; r < 8; ++r) {
        int m = m0 + r + half * 8;     // C/D layout: vgpr r -> rows r / r+8
        float add = (start[m + 1] > start[m]) ? bias : 0.0f;
        out[(size_t)m * DFEAT + n0 + l] = c[r] + add;
    }
}

// ---------------------------------------------------------------------------
extern "C" void kernel_launch(void* const* d_in, const int* in_sizes, int n_in,
                              void* d_out, int out_size, void* d_ws, size_t ws_size,
                              hipStream_t stream) {
    const float* feat = (const float*)d_in[0];
    const int*   ids  = (const int*)d_in[1];
    const float* Wg   = (const float*)d_in[2];
    const float* Wf   = (const float*)d_in[3];
    const float* bf   = (const float*)d_in[4];
    // d_in[5] = num_segments (device scalar); B is derived host-side instead.

    const int N = in_sizes[1];          // 1,000,000 rows
    const int H = in_sizes[4];          // 256
    const int B = out_size / H;         // 4096 segments
    float* out = (float*)d_out;

    // workspace layout: g[N] | S[B*256] | start[B+1]
    float* g = (float*)d_ws;
    float* S = g + (size_t)N;
    int* start = (int*)(S + (size_t)B * DFEAT);

    gap_bounds_kernel<<<(B + 256) / 256, 256, 0, stream>>>(ids, start, N, B);
    gap_gate_kernel<<<(N + 7) / 8, 256, 0, stream>>>(feat, Wg, g, N);
    gap_segment_kernel<<<B, 256, 0, stream>>>(feat, g, start, S);

    const int nTilesN = H / 16;                     // 16
    const int tiles = (B / 16) * nTilesN;           // 4096 tiles
    gap_gemm_kernel<<<tiles / 8, 256, 0, stream>>>(S, Wf, bf, start, out, nTilesN);
}